// BahdanauAttention_65592740545097
// MI455X (gfx1250) — compile-verified
//
#include <hip/hip_runtime.h>

#define BB 4
#define TT 128
#define SS 512
#define HH 512
#define LN_EPS 1e-5f
#define LOG2E 1.4426950408889634f

typedef __attribute__((ext_vector_type(2))) float v2f;
typedef __attribute__((ext_vector_type(8))) float v8f;

__device__ __forceinline__ v2f ld2(const float* p) {
    const float2 t = *reinterpret_cast<const float2*>(p);
    v2f r; r.x = t.x; r.y = t.y; return r;
}

// tanh via native v_exp_f32 / v_rcp_f32: tanh(x) = (e^{2x}-1)/(e^{2x}+1)
__device__ __forceinline__ float fast_tanh(float x) {
    float xc = fminf(fmaxf(x, -15.0f), 15.0f);
    float e  = __builtin_amdgcn_exp2f(xc * (2.0f * LOG2E));  // exp(2x)
    return (e - 1.0f) * __builtin_amdgcn_rcpf(e + 1.0f);
}

#define WMMA_F32(A, Bf, C) \
    __builtin_amdgcn_wmma_f32_16x16x4_f32(false, (A), false, (Bf), (short)0, (C), false, false)

// -------------------------------------------------------------------------
// Pipelined K-strip accumulation for C += X * W^T over Klen (multiple of 16).
// xrow/w* already offset by lane (row base + 2*hi). Double-buffered in regs
// so loads of block i+1 overlap the 16 WMMAs of block i (coarse s_wait).
// -------------------------------------------------------------------------
__device__ __forceinline__ void gemm_strip(const float* __restrict__ xrow,
                                           const float* __restrict__ w0,
                                           const float* __restrict__ w1,
                                           const float* __restrict__ w2,
                                           const float* __restrict__ w3,
                                           int Klen,
                                           v8f& acc0, v8f& acc1, v8f& acc2, v8f& acc3) {
    v2f a_c[4], bc0[4], bc1[4], bc2[4], bc3[4];
#pragma unroll
    for (int j = 0; j < 4; ++j) {
        const int o = 4 * j;
        a_c[j] = ld2(xrow + o);
        bc0[j] = ld2(w0 + o); bc1[j] = ld2(w1 + o);
        bc2[j] = ld2(w2 + o); bc3[j] = ld2(w3 + o);
    }
    for (int k = 16; k < Klen; k += 16) {
        v2f a_n[4], bn0[4], bn1[4], bn2[4], bn3[4];
#pragma unroll
        for (int j = 0; j < 4; ++j) {
            const int o = k + 4 * j;
            a_n[j] = ld2(xrow + o);
            bn0[j] = ld2(w0 + o); bn1[j] = ld2(w1 + o);
            bn2[j] = ld2(w2 + o); bn3[j] = ld2(w3 + o);
        }
        __builtin_prefetch(xrow + k + 112, 0, 3);   // global_prefetch_b8, speculative
#pragma unroll
        for (int j = 0; j < 4; ++j) {
            acc0 = WMMA_F32(a_c[j], bc0[j], acc0);
            acc1 = WMMA_F32(a_c[j], bc1[j], acc1);
            acc2 = WMMA_F32(a_c[j], bc2[j], acc2);
            acc3 = WMMA_F32(a_c[j], bc3[j], acc3);
        }
#pragma unroll
        for (int j = 0; j < 4; ++j) {
            a_c[j] = a_n[j];
            bc0[j] = bn0[j]; bc1[j] = bn1[j];
            bc2[j] = bn2[j]; bc3[j] = bn3[j];
        }
    }
#pragma unroll
    for (int j = 0; j < 4; ++j) {
        acc0 = WMMA_F32(a_c[j], bc0[j], acc0);
        acc1 = WMMA_F32(a_c[j], bc1[j], acc1);
        acc2 = WMMA_F32(a_c[j], bc2[j], acc2);
        acc3 = WMMA_F32(a_c[j], bc3[j], acc3);
    }
}

// -------------------------------------------------------------------------
// C[M x N] = X[M x K] * W[N x K]^T   (row-major). One wave = 16x64 tile.
// -------------------------------------------------------------------------
__global__ void gemm_xwt_kernel(const float* __restrict__ X,
                                const float* __restrict__ W,
                                float* __restrict__ C,
                                int M, int N, int K) {
    const int wave = (blockIdx.x * blockDim.x + threadIdx.x) >> 5;
    const int lane = threadIdx.x & 31;
    const int hi   = lane >> 4;
    const int lr   = lane & 15;
    const int nt   = N >> 6;                // 64-wide N tiles
    const int m0   = (wave / nt) * 16;
    const int n0   = (wave % nt) * 64;

    const float* xrow = X + (size_t)(m0 + lr) * K + 2 * hi;
    const float* w0 = W + (size_t)(n0 +  0 + lr) * K + 2 * hi;
    const float* w1 = W + (size_t)(n0 + 16 + lr) * K + 2 * hi;
    const float* w2 = W + (size_t)(n0 + 32 + lr) * K + 2 * hi;
    const float* w3 = W + (size_t)(n0 + 48 + lr) * K + 2 * hi;

    v8f acc0 = {}, acc1 = {}, acc2 = {}, acc3 = {};
    gemm_strip(xrow, w0, w1, w2, w3, K, acc0, acc1, acc2, acc3);

#pragma unroll
    for (int r = 0; r < 8; ++r) {
        const size_t row = (size_t)(m0 + r + 8 * hi) * N + n0 + lr;
        C[row +  0] = acc0[r];
        C[row + 16] = acc1[r];
        C[row + 32] = acc2[r];
        C[row + 48] = acc3[r];
    }
}

// -------------------------------------------------------------------------
// Fused additive score + mask + softmax. One block (256 thr) per (b,t) row.
// attn[b,t,s] = softmax_s( mask( sum_h tanh(qp[b,t,h]+kp[b,s,h]) * v[h] ) )
// -------------------------------------------------------------------------
__global__ void score_softmax_kernel(const float* __restrict__ qp,
                                     const float* __restrict__ kp,
                                     const float* __restrict__ vvec,
                                     const int* __restrict__ lens,
                                     float* __restrict__ attn) {
    const int b = blockIdx.x / TT;
    const int t = blockIdx.x % TT;
    const int tid  = threadIdx.x;
    const int wv   = tid >> 5;
    const int lane = tid & 31;

    __shared__ __align__(16) float qsh[HH];
    __shared__ __align__(16) float vsh[HH];
    __shared__ float sc[SS];
    __shared__ float redbuf[8];

    for (int i = tid; i < HH; i += 256) {
        qsh[i] = qp[(size_t)(b * TT + t) * HH + i];
        vsh[i] = vvec[i];
    }
    __syncthreads();

    const int len = lens[b];
    const float* kpb = kp + (size_t)b * SS * HH;

    for (int s = wv; s < SS; s += 8) {
        const float* krow = kpb + (size_t)s * HH;
        float acc = 0.0f;
#pragma unroll
        for (int j = 0; j < 4; ++j) {
            const int h = (lane + 32 * j) * 4;          // coalesced float4
            const float4 kq = *(const float4*)(krow + h);
            const float4 qq = *(const float4*)(qsh + h);
            const float4 vq = *(const float4*)(vsh + h);
            acc += fast_tanh(qq.x + kq.x) * vq.x;
            acc += fast_tanh(qq.y + kq.y) * vq.y;
            acc += fast_tanh(qq.z + kq.z) * vq.z;
            acc += fast_tanh(qq.w + kq.w) * vq.w;
        }
#pragma unroll
        for (int o = 16; o >= 1; o >>= 1) acc += __shfl_xor(acc, o, 32);
        if (lane == 0) sc[s] = (s < len) ? acc : -__builtin_inff();
    }
    __syncthreads();

    // block softmax over 512 scores (2 per thread)
    float m = fmaxf(sc[tid], sc[tid + 256]);
#pragma unroll
    for (int o = 16; o >= 1; o >>= 1) m = fmaxf(m, __shfl_xor(m, o, 32));
    if (lane == 0) redbuf[wv] = m;
    __syncthreads();
    float mx = redbuf[0];
#pragma unroll
    for (int i = 1; i < 8; ++i) mx = fmaxf(mx, redbuf[i]);
    __syncthreads();

    const float e0 = __builtin_amdgcn_exp2f((sc[tid]       - mx) * LOG2E);
    const float e1 = __builtin_amdgcn_exp2f((sc[tid + 256] - mx) * LOG2E);
    float ps = e0 + e1;
#pragma unroll
    for (int o = 16; o >= 1; o >>= 1) ps += __shfl_xor(ps, o, 32);
    if (lane == 0) redbuf[wv] = ps;
    __syncthreads();
    float tot = redbuf[0];
#pragma unroll
    for (int i = 1; i < 8; ++i) tot += redbuf[i];
    const float inv = __builtin_amdgcn_rcpf(tot);

    float* arow = attn + (size_t)(b * TT + t) * SS;
    arow[tid]       = e0 * inv;
    arow[tid + 256] = e1 * inv;
}

// -------------------------------------------------------------------------
// context[b] (T x H) = attn[b] (T x S) * enc[b] (S x H). One wave = 16x64.
// B fragments read row-major from enc (coalesced across lanes), pipelined.
// -------------------------------------------------------------------------
__device__ __forceinline__ void ctx_loadB(const float* __restrict__ eb, int ka, int n,
                                          v2f& b0, v2f& b1, v2f& b2, v2f& b3) {
    const float* e0 = eb + (size_t)ka * HH + n;
    b0.x = e0[ 0]; b0.y = e0[HH +  0];
    b1.x = e0[16]; b1.y = e0[HH + 16];
    b2.x = e0[32]; b2.y = e0[HH + 32];
    b3.x = e0[48]; b3.y = e0[HH + 48];
}

__global__ void context_kernel(const float* __restrict__ attn,
                               const float* __restrict__ enc,
                               float* __restrict__ ctx) {
    const int wave = (blockIdx.x * blockDim.x + threadIdx.x) >> 5;
    const int lane = threadIdx.x & 31;
    const int hi   = lane >> 4;
    const int lr   = lane & 15;
    const int b    = wave >> 6;            // 64 waves per batch
    const int rem  = wave & 63;
    const int m0   = (rem >> 3) * 16;      // t tile (T/16 = 8)
    const int n0   = (rem & 7) * 64;       // h tile (H/64 = 8)

    const float* arow = attn + (size_t)(b * TT + m0 + lr) * SS + 2 * hi;
    const float* eb   = enc + (size_t)b * SS * HH;
    const int    nb   = n0 + lr;

    v8f acc0 = {}, acc1 = {}, acc2 = {}, acc3 = {};
    v2f a_c[4], bc0[4], bc1[4], bc2[4], bc3[4];
#pragma unroll
    for (int j = 0; j < 4; ++j) {
        a_c[j] = ld2(arow + 4 * j);
        ctx_loadB(eb, 4 * j + 2 * hi, nb, bc0[j], bc1[j], bc2[j], bc3[j]);
    }
    for (int k = 16; k < SS; k += 16) {
        v2f a_n[4], bn0[4], bn1[4], bn2[4], bn3[4];
#pragma unroll
        for (int j = 0; j < 4; ++j) {
            a_n[j] = ld2(arow + k + 4 * j);
            ctx_loadB(eb, k + 4 * j + 2 * hi, nb, bn0[j], bn1[j], bn2[j], bn3[j]);
        }
#pragma unroll
        for (int j = 0; j < 4; ++j) {
            acc0 = WMMA_F32(a_c[j], bc0[j], acc0);
            acc1 = WMMA_F32(a_c[j], bc1[j], acc1);
            acc2 = WMMA_F32(a_c[j], bc2[j], acc2);
            acc3 = WMMA_F32(a_c[j], bc3[j], acc3);
        }
#pragma unroll
        for (int j = 0; j < 4; ++j) {
            a_c[j] = a_n[j];
            bc0[j] = bn0[j]; bc1[j] = bn1[j];
            bc2[j] = bn2[j]; bc3[j] = bn3[j];
        }
    }
#pragma unroll
    for (int j = 0; j < 4; ++j) {
        acc0 = WMMA_F32(a_c[j], bc0[j], acc0);
        acc1 = WMMA_F32(a_c[j], bc1[j], acc1);
        acc2 = WMMA_F32(a_c[j], bc2[j], acc2);
        acc3 = WMMA_F32(a_c[j], bc3[j], acc3);
    }

#pragma unroll
    for (int r = 0; r < 8; ++r) {
        const size_t row = (size_t)(b * TT + m0 + r + 8 * hi) * HH + n0 + lr;
        ctx[row +  0] = acc0[r];
        ctx[row + 16] = acc1[r];
        ctx[row + 32] = acc2[r];
        ctx[row + 48] = acc3[r];
    }
}

// -------------------------------------------------------------------------
// outp = tanh( [context|query] (M x 2H) * W_out(H x 2H)^T + b_out )
// M = B*T = 512. One wave = 16x64 tile; two K=512 strips (ctx then query).
// -------------------------------------------------------------------------
__global__ void out_gemm_kernel(const float* __restrict__ ctx,
                                const float* __restrict__ query,
                                const float* __restrict__ Wout,
                                const float* __restrict__ bout,
                                float* __restrict__ outp) {
    const int wave = (blockIdx.x * blockDim.x + threadIdx.x) >> 5;
    const int lane = threadIdx.x & 31;
    const int hi   = lane >> 4;
    const int lr   = lane & 15;
    const int m0   = (wave >> 3) * 16;     // (B*T)/16 = 32 tiles
    const int n0   = (wave & 7) * 64;      // H/64 = 8 tiles

    const float* crow = ctx   + (size_t)(m0 + lr) * HH + 2 * hi;
    const float* qrow = query + (size_t)(m0 + lr) * HH + 2 * hi;
    const float* w0 = Wout + (size_t)(n0 +  0 + lr) * (2 * HH) + 2 * hi;
    const float* w1 = Wout + (size_t)(n0 + 16 + lr) * (2 * HH) + 2 * hi;
    const float* w2 = Wout + (size_t)(n0 + 32 + lr) * (2 * HH) + 2 * hi;
    const float* w3 = Wout + (size_t)(n0 + 48 + lr) * (2 * HH) + 2 * hi;

    v8f acc0 = {}, acc1 = {}, acc2 = {}, acc3 = {};
    gemm_strip(crow, w0, w1, w2, w3, HH, acc0, acc1, acc2, acc3);                      // k in [0,H)
    gemm_strip(qrow, w0 + HH, w1 + HH, w2 + HH, w3 + HH, HH, acc0, acc1, acc2, acc3);  // k in [H,2H)

    const float bb0 = bout[n0 +  0 + lr];
    const float bb1 = bout[n0 + 16 + lr];
    const float bb2 = bout[n0 + 32 + lr];
    const float bb3 = bout[n0 + 48 + lr];
#pragma unroll
    for (int r = 0; r < 8; ++r) {
        const size_t row = (size_t)(m0 + r + 8 * hi) * HH + n0 + lr;
        outp[row +  0] = fast_tanh(acc0[r] + bb0);
        outp[row + 16] = fast_tanh(acc1[r] + bb1);
        outp[row + 32] = fast_tanh(acc2[r] + bb2);
        outp[row + 48] = fast_tanh(acc3[r] + bb3);
    }
}

// -------------------------------------------------------------------------
// LayerNorm over last dim (H=512). One wave per row, float4 loads/stores.
// -------------------------------------------------------------------------
__global__ void ln_kernel(const float* __restrict__ x,
                          const float* __restrict__ gamma,
                          const float* __restrict__ beta,
                          float* __restrict__ out) {
    const int wave = (blockIdx.x * blockDim.x + threadIdx.x) >> 5;
    const int lane = threadIdx.x & 31;
    const float* row = x + (size_t)wave * HH;

    float4 val[4];
    float s = 0.0f;
#pragma unroll
    for (int j = 0; j < 4; ++j) {
        val[j] = *(const float4*)(row + (lane + 32 * j) * 4);
        s += val[j].x + val[j].y + val[j].z + val[j].w;
    }
#pragma unroll
    for (int o = 16; o >= 1; o >>= 1) s += __shfl_xor(s, o, 32);
    const float mu = s * (1.0f / HH);

    float q = 0.0f;
#pragma unroll
    for (int j = 0; j < 4; ++j) {
        const float dx = val[j].x - mu, dy = val[j].y - mu;
        const float dz = val[j].z - mu, dw = val[j].w - mu;
        q += dx * dx + dy * dy + dz * dz + dw * dw;
    }
#pragma unroll
    for (int o = 16; o >= 1; o >>= 1) q += __shfl_xor(q, o, 32);
    const float rstd = __builtin_amdgcn_rsqf(q * (1.0f / HH) + LN_EPS);

    float* orow = out + (size_t)wave * HH;
#pragma unroll
    for (int j = 0; j < 4; ++j) {
        const int h = (lane + 32 * j) * 4;
        const float4 g = *(const float4*)(gamma + h);
        const float4 bt = *(const float4*)(beta + h);
        float4 r;
        r.x = (val[j].x - mu) * rstd * g.x + bt.x;
        r.y = (val[j].y - mu) * rstd * g.y + bt.y;
        r.z = (val[j].z - mu) * rstd * g.z + bt.z;
        r.w = (val[j].w - mu) * rstd * g.w + bt.w;
        *(float4*)(orow + h) = r;
    }
}

// -------------------------------------------------------------------------
extern "C" void kernel_launch(void* const* d_in, const int* in_sizes, int n_in,
                              void* d_out, int out_size, void* d_ws, size_t ws_size,
                              hipStream_t stream) {
    const float* query  = (const float*)d_in[0];   // (B,T,H)
    const float* enc    = (const float*)d_in[1];   // (B,S,H)
    const int*   lens   = (const int*)  d_in[2];   // (B,)
    const float* W_h    = (const float*)d_in[3];   // (H,H)
    const float* W_s    = (const float*)d_in[4];   // (H,H)
    const float* vvec   = (const float*)d_in[5];   // (H,)
    const float* W_out  = (const float*)d_in[6];   // (H,2H)
    const float* b_out  = (const float*)d_in[7];   // (H,)
    const float* gamma  = (const float*)d_in[8];   // (H,)
    const float* beta   = (const float*)d_in[9];   // (H,)
    float* out = (float*)d_out;

    float* qp   = (float*)d_ws;                 // B*T*H
    float* kp   = qp   + (size_t)BB * TT * HH;  // B*S*H
    float* attn = kp   + (size_t)BB * SS * HH;  // B*T*S
    float* ctx  = attn + (size_t)BB * TT * SS;  // B*T*H
    float* outp = ctx  + (size_t)BB * TT * HH;  // B*T*H

    // q_proj = query @ W_s^T   : M=512, N=512, K=512 -> 256 waves
    gemm_xwt_kernel<<<32, 256, 0, stream>>>(query, W_s, qp, BB * TT, HH, HH);
    // k_proj = enc @ W_h^T     : M=2048 -> 1024 waves
    gemm_xwt_kernel<<<128, 256, 0, stream>>>(enc, W_h, kp, BB * SS, HH, HH);
    // additive score + mask + softmax
    score_softmax_kernel<<<BB * TT, 256, 0, stream>>>(qp, kp, vvec, lens, attn);
    // context = attn @ enc     : 256 waves
    context_kernel<<<32, 256, 0, stream>>>(attn, enc, ctx);
    // out = tanh([ctx|query] @ W_out^T + b_out) : 256 waves
    out_gemm_kernel<<<32, 256, 0, stream>>>(ctx, query, W_out, b_out, outp);
    // LayerNorm
    ln_kernel<<<64, 256, 0, stream>>>(outp, gamma, beta, out);
}